// LayerConv_70394513981879
// MI455X (gfx1250) — compile-verified
//
#include <hip/hip_runtime.h>
#include <math.h>

// Problem constants (match reference setup_inputs)
#define Nn 50000
#define Bb 4
#define Dd 64
#define Rr 200
#define Qq 64
#define Ee 500000
#define Mm 1000000
#define Ss (Bb * Nn)        // 200000
#define KK (13 * Dd)        // 832
#define XS (KK + 4)         // LDS row stride (padded)

typedef float v2f __attribute__((ext_vector_type(2)));
typedef float v8f __attribute__((ext_vector_type(8)));

// ---------------- init workspace ----------------
__global__ void init_ws(float* __restrict__ sum, float* __restrict__ sq,
                        float* __restrict__ mx, float* __restrict__ mn,
                        int* __restrict__ cnt, float* __restrict__ ssum) {
  int tid = blockIdx.x * 256 + threadIdx.x;
  if (tid < Ss * Dd) {
    sum[tid] = 0.0f;
    sq[tid]  = 0.0f;
    mx[tid]  = -INFINITY;
    mn[tid]  =  INFINITY;
  }
  if (tid < Ss) cnt[tid] = 0;
  if (tid == 0) *ssum = 0.0f;
}

// ---------------- relation = query @ rel_w.T + rel_b (tiny GEMM) ----------------
__global__ void relation_kernel(const float* __restrict__ query,
                                const float* __restrict__ rel_w,
                                const float* __restrict__ rel_b,
                                float* __restrict__ rel) {
  int tid = blockIdx.x * 256 + threadIdx.x;
  if (tid >= Bb * Rr * Dd) return;
  int b   = tid / (Rr * Dd);
  int col = tid % (Rr * Dd);
  const float* q = query + b * Qq;
  const float* w = rel_w + (long)col * Qq;
  float acc = rel_b[col];
#pragma unroll 8
  for (int k = 0; k < Qq; ++k) acc = fmaf(q[k], w[k], acc);
  rel[tid] = acc;  // layout (B, R*D) row-major == (B*R, D)
}

// ---------------- edge scatter: 16 lanes per edge, float4 per lane ----------------
__global__ void scatter_kernel(const float* __restrict__ input,
                               const float* __restrict__ rel,
                               const int* __restrict__ edge_index,
                               const int* __restrict__ edge_type,
                               const int* __restrict__ mask_batch,
                               const int* __restrict__ mask_edge,
                               float* __restrict__ sum, float* __restrict__ sq,
                               float* __restrict__ mx, float* __restrict__ mn,
                               int* __restrict__ cnt) {
  int tid = blockIdx.x * 256 + threadIdx.x;
  if (tid >= Mm * 16) return;
  int e  = tid >> 4;
  int d4 = tid & 15;
  int me  = mask_edge[e];
  int mb  = mask_batch[e];
  int rid = edge_type[me];
  int head = edge_index[me];        // edge_index[0][me]
  int tail = edge_index[Ee + me];   // edge_index[1][me]
  long off_head = (long)head + (long)Nn * mb;
  long off_tail = (long)tail + (long)Nn * mb;

  float4 hin = ((const float4*)(input + off_head * 64))[d4];
  float4 re  = ((const float4*)(rel + ((long)mb * Rr + rid) * 64))[d4];
  float4 msg = make_float4(hin.x * re.x, hin.y * re.y, hin.z * re.z, hin.w * re.w);

  long base = off_tail * 64 + d4 * 4;
  atomicAdd(&sum[base + 0], msg.x);
  atomicAdd(&sum[base + 1], msg.y);
  atomicAdd(&sum[base + 2], msg.z);
  atomicAdd(&sum[base + 3], msg.w);
  atomicAdd(&sq[base + 0], msg.x * msg.x);
  atomicAdd(&sq[base + 1], msg.y * msg.y);
  atomicAdd(&sq[base + 2], msg.z * msg.z);
  atomicAdd(&sq[base + 3], msg.w * msg.w);
  atomicMax(&mx[base + 0], msg.x);
  atomicMax(&mx[base + 1], msg.y);
  atomicMax(&mx[base + 2], msg.z);
  atomicMax(&mx[base + 3], msg.w);
  atomicMin(&mn[base + 0], msg.x);
  atomicMin(&mn[base + 1], msg.y);
  atomicMin(&mn[base + 2], msg.z);
  atomicMin(&mn[base + 3], msg.w);
  if (d4 == 0) atomicAdd(&cnt[off_tail], 1);
}

// ---------------- global sum of log(deg) for scale normalization ----------------
__global__ void logdeg_kernel(const int* __restrict__ cnt,
                              const float* __restrict__ degree_in,
                              float* __restrict__ ssum) {
  int s = blockIdx.x * 256 + threadIdx.x;
  float v = 0.0f;
  if (s < Ss) {
    float deg = ((cnt[s] > 0) ? degree_in[s % Nn] : 0.0f) + 1.0f;
    v = logf(deg);
  }
#pragma unroll
  for (int off = 16; off > 0; off >>= 1) v += __shfl_down(v, off, 32);
  if ((threadIdx.x & 31) == 0) atomicAdd(ssum, v);
}

// ---------------- fused: features -> WMMA GEMM -> LayerNorm -> ReLU ----------------
// One wave32 per 16-row tile. C(16x64) = A(16x832) x B(832x64), f32 WMMA 16x16x4.
__global__ __launch_bounds__(32) void final_kernel(
    const float* __restrict__ input, const float* __restrict__ boundary,
    const float* __restrict__ degree_in,
    const float* __restrict__ lin_w, const float* __restrict__ lin_b,
    const float* __restrict__ ln_g, const float* __restrict__ ln_b,
    const float* __restrict__ agg_sum, const float* __restrict__ sq_sum,
    const float* __restrict__ agg_max, const float* __restrict__ agg_min,
    const int* __restrict__ cnt, const float* __restrict__ ssum,
    float* __restrict__ out) {
  __shared__ float xs[16 * XS];      // 16 x 832 feature tile (padded)
  __shared__ float hbuf[16 * 64];    // GEMM result tile
  __shared__ float scs[16], invs[16], invdeg[16], mu_s[16], rs_s[16];
  __shared__ int cnts[16];

  const int lane = threadIdx.x;
  const int s0   = blockIdx.x * 16;
  const int l16  = lane & 15;
  const int half = lane >> 4;

  // per-row scalars
  if (lane < 16) {
    int s = s0 + lane;
    int c = cnt[s];
    float deg = ((c > 0) ? degree_in[s % Nn] : 0.0f) + 1.0f;
    float mean_scale = *ssum / (float)Ss;
    float sc = logf(deg) / mean_scale;
    scs[lane]    = sc;
    invs[lane]   = 1.0f / fmaxf(sc, 0.01f);
    invdeg[lane] = 1.0f / deg;
    cnts[lane]   = c;
  }
  __syncthreads();

  // build feature rows: x = [input(64) | {mean,max,min,std}_d x {1,sc,1/sc}]
  {
    int r = l16;
    int s = s0 + r;
    float sc = scs[r], iv = invs[r], idg = invdeg[r];
    bool has = cnts[r] > 0;
    int d0 = half * 32;
    for (int d = d0; d < d0 + 32; ++d) {
      long gi = (long)s * 64 + d;
      float xin = input[gi];
      xs[r * XS + d] = xin;
      float bnd  = boundary[gi];
      float mean = (agg_sum[gi] + bnd) * idg;
      float mxv  = has ? agg_max[gi] : 0.0f;
      float mnv  = has ? agg_min[gi] : 0.0f;
      float stdv = fmaxf(sq_sum[gi] + bnd * bnd, 1e-6f) * idg;
      float st[4] = {mean, mxv, mnv, stdv};
      int base = r * XS + 64 + d * 12;
#pragma unroll
      for (int t = 0; t < 4; ++t) {
        float v = st[t];
        xs[base + t * 3 + 0] = v;
        xs[base + t * 3 + 1] = v * sc;
        xs[base + t * 3 + 2] = v * iv;
      }
    }
  }
  __syncthreads();

  // WMMA GEMM: 4 N-tiles, K-loop step 4
  v8f c0 = {}, c1 = {}, c2 = {}, c3 = {};
  const float* arow = &xs[l16 * XS + 2 * half];
  const float* b0p = lin_w + (long)(0  + l16) * KK + 2 * half;
  const float* b1p = lin_w + (long)(16 + l16) * KK + 2 * half;
  const float* b2p = lin_w + (long)(32 + l16) * KK + 2 * half;
  const float* b3p = lin_w + (long)(48 + l16) * KK + 2 * half;
#pragma unroll 4
  for (int k0 = 0; k0 < KK; k0 += 4) {
    v2f a  = *(const v2f*)(arow + k0);
    v2f b0 = *(const v2f*)(b0p + k0);
    v2f b1 = *(const v2f*)(b1p + k0);
    v2f b2 = *(const v2f*)(b2p + k0);
    v2f b3 = *(const v2f*)(b3p + k0);
    c0 = __builtin_amdgcn_wmma_f32_16x16x4_f32(false, a, false, b0, (short)0, c0, false, false);
    c1 = __builtin_amdgcn_wmma_f32_16x16x4_f32(false, a, false, b1, (short)0, c1, false, false);
    c2 = __builtin_amdgcn_wmma_f32_16x16x4_f32(false, a, false, b2, (short)0, c2, false, false);
    c3 = __builtin_amdgcn_wmma_f32_16x16x4_f32(false, a, false, b3, (short)0, c3, false, false);
  }

  // bias + stash to LDS (C layout: VGPR j -> M=j+8*half, N=nt*16+l16)
  float bia0 = lin_b[0  + l16];
  float bia1 = lin_b[16 + l16];
  float bia2 = lin_b[32 + l16];
  float bia3 = lin_b[48 + l16];
#pragma unroll
  for (int j = 0; j < 8; ++j) {
    int m = j + 8 * half;
    hbuf[m * 64 + 0  + l16] = c0[j] + bia0;
    hbuf[m * 64 + 16 + l16] = c1[j] + bia1;
    hbuf[m * 64 + 32 + l16] = c2[j] + bia2;
    hbuf[m * 64 + 48 + l16] = c3[j] + bia3;
  }
  __syncthreads();

  // LayerNorm stats per row
  if (lane < 16) {
    float su = 0.0f, s2 = 0.0f;
    for (int c = 0; c < 64; ++c) {
      float v = hbuf[lane * 64 + c];
      su += v; s2 += v * v;
    }
    float mu  = su * (1.0f / 64.0f);
    float var = s2 * (1.0f / 64.0f) - mu * mu;
    mu_s[lane] = mu;
    rs_s[lane] = rsqrtf(var + 1e-5f);
  }
  __syncthreads();

  // normalize + affine + relu + store
#pragma unroll 4
  for (int i = 0; i < 32; ++i) {
    int j = lane + 32 * i;
    int r = j >> 6, c = j & 63;
    float v = (hbuf[j] - mu_s[r]) * rs_s[r] * ln_g[c] + ln_b[c];
    out[(long)s0 * 64 + j] = fmaxf(v, 0.0f);
  }
}

extern "C" void kernel_launch(void* const* d_in, const int* in_sizes, int n_in,
                              void* d_out, int out_size, void* d_ws, size_t ws_size,
                              hipStream_t stream) {
  (void)in_sizes; (void)n_in; (void)out_size; (void)ws_size;
  const float* input     = (const float*)d_in[0];
  const float* query     = (const float*)d_in[1];
  const float* boundary  = (const float*)d_in[2];
  const float* degree_in = (const float*)d_in[3];
  const int*   edge_index = (const int*)d_in[4];
  const int*   edge_type  = (const int*)d_in[5];
  const int*   mask_batch = (const int*)d_in[6];
  const int*   mask_edge  = (const int*)d_in[7];
  const float* rel_w = (const float*)d_in[8];
  const float* rel_b = (const float*)d_in[9];
  const float* lin_w = (const float*)d_in[10];
  const float* lin_b = (const float*)d_in[11];
  const float* ln_g  = (const float*)d_in[12];
  const float* ln_b  = (const float*)d_in[13];
  float* out = (float*)d_out;

  // workspace layout (floats)
  float* rel = (float*)d_ws;                          // B*R*D
  float* sum = rel + (size_t)Bb * Rr * Dd;            // S*D
  float* sq  = sum + (size_t)Ss * Dd;                 // S*D
  float* mx  = sq  + (size_t)Ss * Dd;                 // S*D
  float* mn  = mx  + (size_t)Ss * Dd;                 // S*D
  int*   cnt = (int*)(mn + (size_t)Ss * Dd);          // S
  float* ssum = (float*)(cnt + Ss);                   // 1

  init_ws<<<(Ss * Dd + 255) / 256, 256, 0, stream>>>(sum, sq, mx, mn, cnt, ssum);
  relation_kernel<<<(Bb * Rr * Dd + 255) / 256, 256, 0, stream>>>(query, rel_w, rel_b, rel);
  scatter_kernel<<<(Mm * 16 + 255) / 256, 256, 0, stream>>>(
      input, rel, edge_index, edge_type, mask_batch, mask_edge, sum, sq, mx, mn, cnt);
  logdeg_kernel<<<(Ss + 255) / 256, 256, 0, stream>>>(cnt, degree_in, ssum);
  final_kernel<<<Ss / 16, 32, 0, stream>>>(
      input, boundary, degree_in, lin_w, lin_b, ln_g, ln_b,
      sum, sq, mx, mn, cnt, ssum, out);
}